// Summary_model_38448547234395
// MI455X (gfx1250) — compile-verified
//
#include <hip/hip_runtime.h>
#include <hip/hip_bf16.h>
#include <math.h>

typedef float v2f __attribute__((ext_vector_type(2)));
typedef float v4f __attribute__((ext_vector_type(4)));
typedef float v8f __attribute__((ext_vector_type(8)));
typedef int   v4i __attribute__((ext_vector_type(4)));

#define DIM_SENT 256
#define DIM_CLU  128
#define MAXSEG   128   // k+1 must be <= 128 (k=100 here)

// ---------------------------------------------------------------------------
// Kernel 1: prep — weight-norm, scaled sentence weights, cluster dot terms,
// zero cluster sums. One block of 256 threads; trivially cheap.
// ---------------------------------------------------------------------------
__global__ void prep_kernel(const float* __restrict__ v,
                            const float* __restrict__ g,
                            const float* __restrict__ clu,
                            int k,
                            float* __restrict__ W,    // [256]
                            float* __restrict__ CD,   // [k+1], CD[0]=0
                            float* __restrict__ CS) { // [k+1] zeroed
  __shared__ float red[256];
  __shared__ float s_scale;
  const int t = threadIdx.x;

  // ||v||^2 over 384 elements
  float x = v[t];
  float acc = x * x;
  if (t < 128) { float y = v[256 + t]; acc += y * y; }
  red[t] = acc;
  __syncthreads();
  for (int s = 128; s > 0; s >>= 1) {
    if (t < s) red[t] += red[t + s];
    __syncthreads();
  }
  if (t == 0) s_scale = g[0] / sqrtf(red[0]);
  __syncthreads();
  const float scale = s_scale;

  // scaled sentence weights
  W[t] = v[t] * scale;

  // zero cluster sums; CD[0]=0 (noise segment)
  if (t <= k) CS[t] = 0.0f;
  if (t == 0) CD[0] = 0.0f;

  // per-cluster dot with w[256:384]  (100 x 128 — tiny)
  if (t < k) {
    const float* row = clu + (size_t)t * DIM_CLU;
    float a = 0.0f;
    #pragma unroll 8
    for (int j = 0; j < DIM_CLU; ++j) a += row[j] * v[DIM_SENT + j];
    CD[t + 1] = a * scale;
  }
}

// ---------------------------------------------------------------------------
// Kernel 2: score — bandwidth-bound matvec via V_WMMA_F32_16X16X4_F32.
//
// Tile = 16 sentences per wave iteration. A (16x4 f32) layout: lanes 0-15 =
// rows M, lanes 16-31 = rows M again; VGPR0/1 = K pair of the lane half.
// B rows hold W broadcast into all 16 columns, so D[m][*] = dot(row m, W).
// Using the K-permutation freedom, each lane streams its row with one
// non-temporal global_load_b128 at offset 8t + 4h (h = lane>=16), matched by
// W from LDS, feeding two WMMAs per iteration. C accumulates over all 256
// dims. Sentence data (512 MB, single-use, > 192 MB L2) is loaded NT so it
// does not evict reused lines (labels, weights, scores).
// Epilogue: lanes 0 and 16 hold dots for sentences base+8h+0..7 in C[0..7].
// ---------------------------------------------------------------------------
__global__ void __launch_bounds__(256)
score_kernel(const float* __restrict__ sent,
             const int*   __restrict__ labels,
             const float* __restrict__ bptr,
             const float* __restrict__ W,
             const float* __restrict__ CD,
             float*       __restrict__ CS,
             float*       __restrict__ SC,
             int n, int tiles, int k) {
  __shared__ float lw[DIM_SENT];   // weights
  __shared__ float lcd[MAXSEG];    // cluster dot terms (by seg)
  __shared__ float lcs[MAXSEG];    // block-local cluster sums

  const int t = threadIdx.x;
  lw[t] = W[t];
  if (t <= k) { lcd[t] = CD[t]; lcs[t] = 0.0f; }
  __syncthreads();

  const float bias = bptr[0];
  const int lane  = t & 31;
  const int wave  = t >> 5;
  const int h     = lane >> 4;       // lane half: which K sub-pair we stream
  const int mrow  = lane & 15;       // row within tile
  const int gwave = blockIdx.x * 8 + wave;
  const int nwav  = gridDim.x * 8;

  for (int T = gwave; T < tiles; T += nwav) {
    const int base = T * 16;
    int row = base + mrow;
    row = (row < n) ? row : (n - 1);          // branchless tail clamp
    // v4f index: (8t + 4h)/4 = 2t + h
    const v4f* rp = (const v4f*)(sent + (size_t)row * DIM_SENT) + h;
    const v4f* wp = (const v4f*)(lw) + h;

    v8f c = {};
    #pragma unroll 4
    for (int it = 0; it < DIM_SENT / 8; ++it) {
      const v4f aa = __builtin_nontemporal_load(rp + 2 * it); // elems 8it+4h..+3
      const v4f ww = wp[2 * it];                              // W same offsets
      v2f A0 = {aa[0], aa[1]}; v2f B0 = {ww[0], ww[1]};
      v2f A1 = {aa[2], aa[3]}; v2f B1 = {ww[2], ww[3]};
      c = __builtin_amdgcn_wmma_f32_16x16x4_f32(false, A0, false, B0,
                                                (short)0, c, false, false);
      c = __builtin_amdgcn_wmma_f32_16x16x4_f32(false, A1, false, B1,
                                                (short)0, c, false, false);
    }

    // C/D layout: lane L<16 VGPR r = D[M=r][N=L]; lane L>=16 VGPR r = D[M=r+8].
    // All N columns equal -> lane 0 has dots 0..7, lane 16 has dots 8..15.
    if (mrow == 0) {
      const int rbase = base + h * 8;
      if (rbase + 8 <= n) {
        // fast path: two b128 label loads (rbase is a multiple of 8 -> aligned)
        const v4i lab0 = *(const v4i*)(labels + rbase);
        const v4i lab1 = *(const v4i*)(labels + rbase + 4);
        #pragma unroll
        for (int r = 0; r < 8; ++r) {
          const int seg = ((r < 4) ? lab0[r & 3] : lab1[r & 3]) + 1; // 0..k
          const float s = tanhf(c[r] + lcd[seg] + bias);
          SC[rbase + r] = s;
          atomicAdd(&lcs[seg], s);             // ds_add_f32
        }
      } else {
        #pragma unroll
        for (int r = 0; r < 8; ++r) {
          const int i = rbase + r;
          if (i < n) {
            const int seg = labels[i] + 1;
            const float s = tanhf(c[r] + lcd[seg] + bias);
            SC[i] = s;
            atomicAdd(&lcs[seg], s);
          }
        }
      }
    }
  }

  __syncthreads();
  if (t <= k) {
    const float val = lcs[t];
    if (val != 0.0f) atomicAdd(&CS[t], val);
  }
}

// ---------------------------------------------------------------------------
// Kernel 3: finalize — salience normalization + position score blend.
// ---------------------------------------------------------------------------
__global__ void finalize_kernel(const int*   __restrict__ labels,
                                const float* __restrict__ SC,
                                const float* __restrict__ CS,
                                float*       __restrict__ out,
                                int n, float invcb) {
  const int i = blockIdx.x * blockDim.x + threadIdx.x;
  if (i >= n) return;
  const int lab = labels[i];
  float sal = 0.0f;
  if (lab >= 0) sal = SC[i] / CS[lab + 1];
  const float pos = fmaxf(0.5f, expf(-(float)(i + 1) * invcb));
  out[i] = 0.5f * sal + 0.5f * pos;
}

// ---------------------------------------------------------------------------
// Host launcher
// ---------------------------------------------------------------------------
extern "C" void kernel_launch(void* const* d_in, const int* in_sizes, int n_in,
                              void* d_out, int out_size, void* d_ws, size_t ws_size,
                              hipStream_t stream) {
  const int*   labels = (const int*)d_in[0];
  const float* sent   = (const float*)d_in[1];
  const float* clu    = (const float*)d_in[2];
  const float* v      = (const float*)d_in[3];
  const float* g      = (const float*)d_in[4];
  const float* b      = (const float*)d_in[5];

  const int n = in_sizes[0];
  const int k = in_sizes[2] / DIM_CLU;

  // workspace layout (floats):
  //   [0,256)      W    scaled sentence weights
  //   [256,384)    CD   cluster dot terms indexed by seg (CD[0]=0)
  //   [384,512)    CS   cluster score sums
  //   [512,512+n)  SC   per-sentence scores
  float* ws = (float*)d_ws;
  float* W  = ws;
  float* CD = ws + 256;
  float* CS = ws + 384;
  float* SC = ws + 512;

  prep_kernel<<<1, 256, 0, stream>>>(v, g, clu, k, W, CD, CS);

  const int tiles = (n + 15) / 16;
  int blocks = (tiles + 31) / 32;       // ~4 tiles per wave at 8 waves/block
  if (blocks > 1024) blocks = 1024;
  if (blocks < 1)    blocks = 1;
  score_kernel<<<blocks, 256, 0, stream>>>(sent, labels, b, W, CD, CS, SC,
                                           n, tiles, k);

  const float invcb = 1.0f / cbrtf((float)n);
  finalize_kernel<<<(n + 255) / 256, 256, 0, stream>>>(labels, SC, CS,
                                                       (float*)d_out, n, invcb);
}